// FeatureExtractor_33303176413506
// MI455X (gfx1250) — compile-verified
//
#include <hip/hip_runtime.h>
#include <math.h>

typedef __attribute__((ext_vector_type(2))) float v2f;
typedef __attribute__((ext_vector_type(8))) float v8f;

#define HOPX      176
#define FLEN      704
#define HALFX     352
#define NFRAMES   498
#define NBATCH    2
#define AUDIO_T   88200
#define THRESH    0.1f

#define APAD      240              // basepad index offset (j = idx - APAD)
#define ABUF      832              // 240 + 592
#define BBUF      864              // max B index 862, rounded up
#define KTOT      592              // K extent of the banded matmul

// One wave32 per (batch, frame). YIN difference function via f32 WMMA
// (exact precision match to the f32 reference), then CMND + search.
// Operand masking is baked into zero-padded LDS images so the inner loop
// is branch-free: ds_load_b64/b32 + back-to-back v_wmma only.
__global__ __launch_bounds__(32)
void yin_wmma_kernel(const float* __restrict__ audio, float* __restrict__ out)
{
    __shared__ __align__(16) float basepad[ABUF]; // A image: x[0:352] zero-padded both sides
    __shared__ __align__(16) float fullpad[BBUF]; // B image: x[0:704] + zero tail
    __shared__ float P[FLEN + 1];                 // exclusive prefix sum of x^2
    __shared__ float diffs[HALFX];
    __shared__ float cmnd[HALFX];

    const int lane = threadIdx.x;                 // 0..31 (wave32)
    const int bf   = blockIdx.x;
    const int b    = bf / NFRAMES;
    const int f    = bf - b * NFRAMES;
    const float* src = audio + b * AUDIO_T + f * HOPX;

    // ---- 1. build zero-padded LDS images ----
    for (int i = lane; i < BBUF; i += 32)
        fullpad[i] = (i < FLEN) ? src[i] : 0.f;
    __syncthreads();
    for (int j = lane; j < ABUF; j += 32)
        basepad[j] = (j >= APAD && j < APAD + HALFX) ? fullpad[j - APAD] : 0.f;

    // ---- 2. exclusive prefix sum of squares: P[i] = sum_{k<i} x[k]^2 ----
    {
        float loc[22];
        float run = 0.f;
        const int base = lane * 22;
        #pragma unroll
        for (int k = 0; k < 22; ++k) {
            float v = fullpad[base + k];
            run += v * v;
            loc[k] = run;
        }
        float incl = run;
        #pragma unroll
        for (int off = 1; off < 32; off <<= 1) {
            float n = __shfl_up(incl, off, 32);
            if (lane >= off) incl += n;
        }
        const float excl = incl - run;
        #pragma unroll
        for (int k = 0; k < 22; ++k) P[base + k + 1] = excl + loc[k];
        if (lane == 0) P[0] = 0.f;
    }
    __syncthreads();

    const float E0      = P[HALFX];
    const float totalSq = P[FLEN];

    // Per-lane WMMA operand coordinates (ISA 32-bit A / B / C-D layouts):
    //   A 16x4:  lanes 0-15 -> M=lane, K={0,1}; lanes 16-31 -> M=lane-16, K={2,3}
    //   D 16x16: VGPR v -> row M=v (lanes 0-15) / M=v+8 (lanes 16-31), N=lane&15
    const int halfw = lane >> 4;
    const int l15   = lane & 15;
    const int kq    = halfw * 2;

    // A index is even (APAD, kq, 16*l15, m0 all even) -> aligned v2f (b64) load.
    const int aoff = APAD + kq - 16 * l15;   // >= 0
    const int boff = kq + l15;               // tile0 B base; tile1 adds +256

    // ---- 3. fused banded matmul: both tau-tiles share each A load ----
    v8f acc0 = {0.f, 0.f, 0.f, 0.f, 0.f, 0.f, 0.f, 0.f};
    v8f acc1 = acc0;
    #pragma unroll 8
    for (int m0 = 0; m0 < KTOT; m0 += 4) {
        const v2f a = *reinterpret_cast<const v2f*>(&basepad[aoff + m0]);
        v2f b0, b1;
        b0.x = fullpad[boff + m0];
        b0.y = fullpad[boff + m0 + 1];
        b1.x = fullpad[boff + m0 + 256];
        b1.y = fullpad[boff + m0 + 257];
        acc0 = __builtin_amdgcn_wmma_f32_16x16x4_f32(
                   false, a, false, b0, (short)0, acc0, false, false);
        acc1 = __builtin_amdgcn_wmma_f32_16x16x4_f32(
                   false, a, false, b1, (short)0, acc1, false, false);
    }

    // ---- 4. diff(tau) = E(0) + E(tau) - 2*r(tau) ----
    #pragma unroll
    for (int v = 0; v < 8; ++v) {
        const int M   = v + 8 * halfw;
        const int tau = 16 * M + l15;                 // tile 0: 0..255
        const float Et = P[tau + HALFX] - P[tau];
        diffs[tau] = E0 + Et - 2.f * acc0[v];
    }
    #pragma unroll
    for (int v = 0; v < 8; ++v) {
        const int M   = v + 8 * halfw;
        const int tau = 256 + 16 * M + l15;           // tile 1: keep 256..351
        if (tau < HALFX) {
            const float Et = P[tau + HALFX] - P[tau];
            diffs[tau] = E0 + Et - 2.f * acc1[v];
        }
    }
    __syncthreads();

    // ---- 5. cumsum + CMND ----
    float myLast;
    {
        float loc[11];
        float run = 0.f;
        const int base = lane * 11;
        #pragma unroll
        for (int k = 0; k < 11; ++k) {
            run += diffs[base + k];
            loc[k] = run;
        }
        float incl = run;
        #pragma unroll
        for (int off = 1; off < 32; off <<= 1) {
            float n = __shfl_up(incl, off, 32);
            if (lane >= off) incl += n;
        }
        const float excl = incl - run;
        #pragma unroll
        for (int k = 0; k < 11; ++k) {
            const int tau = base + k;
            const float cum = excl + loc[k];
            float c;
            if (tau == 0)       c = 0.f;
            else if (cum > 0.f) c = diffs[tau] * (float)tau / cum;
            else                c = 1.f;
            cmnd[tau] = c;
        }
        myLast = excl + loc[10];                      // cums[351] in lane 31
    }
    const float cumsTotal = __shfl(myLast, 31, 32);
    __syncthreads();

    // ---- 6. first tau in [2,350] with cmnd<thr and cmnd[tau]<cmnd[tau+1] ----
    int best = 1 << 30;
    for (int tau = 2 + lane; tau <= 350; tau += 32) {
        const float c = cmnd[tau];
        if (c < THRESH && c < cmnd[tau + 1]) { best = tau; break; }
    }
    #pragma unroll
    for (int off = 16; off > 0; off >>= 1) {
        const int o = __shfl_xor(best, off, 32);
        best = (o < best) ? o : best;
    }

    // ---- 7. finalize on lane 0 ----
    if (lane == 0) {
        const bool found       = best < (1 << 30);
        const bool unvoicedAll = (cumsTotal - diffs[0]) == 0.f;
        float f0v = 0.f;
        if (found && !unvoicedAll) {
            const float f0hz = 44100.f / (float)best;
            if (f0hz >= 32.7f && f0hz <= 2093.f) {
                const float lo = log2f(32.7f);
                const float hi = log2f(2093.f);
                f0v = (log2f(f0hz) - lo) / (hi - lo);
            }
        }
        float loud = sqrtf(totalSq / (float)FLEN) / 0.3f;
        loud = loud < 0.f ? 0.f : (loud > 1.f ? 1.f : loud);
        out[b * NFRAMES + f]                    = f0v;   // f0 block
        out[NBATCH * NFRAMES + b * NFRAMES + f] = loud;  // loudness block
    }
}

extern "C" void kernel_launch(void* const* d_in, const int* in_sizes, int n_in,
                              void* d_out, int out_size, void* d_ws, size_t ws_size,
                              hipStream_t stream)
{
    (void)in_sizes; (void)n_in; (void)out_size; (void)d_ws; (void)ws_size;
    const float* audio = (const float*)d_in[0];
    float* out = (float*)d_out;
    hipLaunchKernelGGL(yin_wmma_kernel,
                       dim3(NBATCH * NFRAMES), dim3(32), 0, stream,
                       audio, out);
}